// PostProcess_51539607776
// MI455X (gfx1250) — compile-verified
//
#include <hip/hip_runtime.h>
#include <hip/hip_bf16.h>
#include <stdint.h>

// ---------------- problem constants ----------------
constexpr int B_    = 16;
constexpr int Q_    = 2304;
constexpr int C_    = 1204;
constexpr int NPB   = Q_ * C_;        // 2,774,016 elements per batch
constexpr int KTOP  = 100;

constexpr int TILE_F  = 512;          // floats per async tile (2 KB)
constexpr int NTILES  = NPB / TILE_F; // 5418 (exact)
constexpr int IPT     = (TILE_F * 4) / 512; // 4 x b128x32lane per tile
constexpr int NBINS   = 4096;         // key[31:20]
constexpr int CAP     = 4096;         // candidate cap per batch
constexpr int THREADS = 256;          // 8 waves (wave32)
constexpr int WAVES   = THREADS / 32;
constexpr int P1_BLOCKS = 32;         // blocks per batch for streaming passes

// ---------------- helpers ----------------
__device__ __forceinline__ unsigned mono(float f) {
  unsigned u = __float_as_uint(f);
  return (u & 0x80000000u) ? ~u : (u | 0x80000000u);
}
__device__ __forceinline__ float unmono(unsigned u) {
  unsigned b = (u & 0x80000000u) ? (u & 0x7FFFFFFFu) : ~u;
  return __uint_as_float(b);
}
__device__ __forceinline__ unsigned lds_off(const void* p) {
  // generic LDS pointer: high dword = shared aperture, low dword = LDS byte offset
  return (unsigned)(uintptr_t)p;
}
// CDNA5 async global->LDS copy: per-lane 16B, tracked by ASYNCcnt
__device__ __forceinline__ void async_b128(unsigned ldsoff, unsigned long long gaddr) {
  asm volatile("global_load_async_to_lds_b128 %0, %1, off"
               :: "v"(ldsoff), "v"(gaddr) : "memory");
}
__device__ __forceinline__ void wait_async_half() {
  asm volatile("s_wait_asynccnt 4" ::: "memory");
}
__device__ __forceinline__ void wait_async_zero() {
  asm volatile("s_wait_asynccnt 0" ::: "memory");
}

__device__ __forceinline__ unsigned make_key(float x, const float* __restrict__ obj,
                                             int b, int e) {
  int q = e / C_;
  int c = e - q * C_;
  if (c == C_ - 1) {
    float p = obj[b * Q_ + q] * (1.0f / (1.0f + __expf(-x)));
    x = __logf(p / (1.0f - p));   // adjusted logit: sigmoid(x') == sigmoid(x)*obj
  }
  return mono(x);
}

// ---------------- pass 1: per-batch histogram over key[31:20] ----------------
__global__ __launch_bounds__(THREADS) void k_hist(const float* __restrict__ logits,
                                                  const float* __restrict__ obj,
                                                  unsigned* __restrict__ hist) {
  __shared__ unsigned s_hist[NBINS];                                   // 16 KB
  __shared__ __align__(16) unsigned char s_stage[WAVES * 2 * TILE_F * 4]; // 32 KB
  const int b = blockIdx.y;
  const int tid = threadIdx.x;
  for (int i = tid; i < NBINS; i += THREADS) s_hist[i] = 0u;
  __syncthreads();

  const int wave = tid >> 5, lane = tid & 31;
  const int wstride = P1_BLOCKS * WAVES;
  const unsigned long long gbase =
      (unsigned long long)(uintptr_t)logits + (unsigned long long)b * NPB * 4ull;
  unsigned char* mybuf = s_stage + wave * (2 * TILE_F * 4);
  const unsigned lbase = lds_off(mybuf);

  auto issue = [&](int t, int sel) {
    unsigned lo = lbase + (unsigned)sel * (TILE_F * 4) + (unsigned)lane * 16u;
    unsigned long long ga = gbase + (unsigned long long)t * (TILE_F * 4) + (unsigned)lane * 16u;
#pragma unroll
    for (int j = 0; j < IPT; ++j) { async_b128(lo, ga); lo += 512u; ga += 512ull; }
  };

  int t = blockIdx.x * WAVES + wave;
  int cur = 0;
  if (t < NTILES) issue(t, 0);
  for (; t < NTILES; t += wstride) {
    const int tn = t + wstride;
    if (tn < NTILES) { issue(tn, cur ^ 1); wait_async_half(); }
    else             { wait_async_zero(); }
    const float4* bp = reinterpret_cast<const float4*>(mybuf + cur * (TILE_F * 4));
#pragma unroll
    for (int k = 0; k < IPT; ++k) {
      float4 v = bp[k * 32 + lane];
      const int e0 = t * TILE_F + k * 128 + lane * 4;
      float vals[4] = {v.x, v.y, v.z, v.w};
#pragma unroll
      for (int m = 0; m < 4; ++m) {
        unsigned key = make_key(vals[m], obj, b, e0 + m);
        atomicAdd(&s_hist[key >> 20], 1u);
      }
    }
    cur ^= 1;
  }
  __syncthreads();
  for (int i = tid; i < NBINS; i += THREADS) {
    unsigned c = s_hist[i];
    if (c) atomicAdd(&hist[b * NBINS + i], c);
  }
}

// ---------------- threshold: suffix-scan histogram, find top-K crossing ----------------
__global__ __launch_bounds__(256) void k_thresh(const unsigned* __restrict__ hist,
                                                unsigned* __restrict__ thresh) {
  __shared__ unsigned s_part[256];
  const int b = blockIdx.x, t = threadIdx.x;
  const unsigned* h = hist + b * NBINS;
  unsigned seg[16];
  unsigned local = 0;
#pragma unroll
  for (int i = 0; i < 16; ++i) { seg[i] = h[t * 16 + i]; local += seg[i]; }
  s_part[t] = local;
  __syncthreads();
  // inclusive suffix scan (Hillis-Steele)
  for (int off = 1; off < 256; off <<= 1) {
    unsigned v = (t + off < 256) ? s_part[t + off] : 0u;
    __syncthreads();
    s_part[t] += v;
    __syncthreads();
  }
  const unsigned sufIncl  = s_part[t];
  const unsigned sufAfter = sufIncl - local;
  if (sufIncl >= (unsigned)KTOP && sufAfter < (unsigned)KTOP) {
    unsigned run = sufAfter;
    int bin = 0;
    for (int i = 15; i >= 0; --i) {
      run += seg[i];
      if (run >= (unsigned)KTOP) { bin = t * 16 + i; break; }
    }
    thresh[b] = (unsigned)bin << 20;   // lower edge of threshold bin
  }
  if (t == 0 && s_part[0] < (unsigned)KTOP) thresh[b] = 0u; // degenerate: accept all
}

// ---------------- pass 2: collect candidates >= threshold ----------------
__global__ __launch_bounds__(THREADS) void k_collect(const float* __restrict__ logits,
                                                     const float* __restrict__ obj,
                                                     const unsigned* __restrict__ thresh,
                                                     unsigned* __restrict__ ccnt,
                                                     unsigned* __restrict__ ckey,
                                                     unsigned* __restrict__ cidx) {
  __shared__ __align__(16) unsigned char s_stage[WAVES * 2 * TILE_F * 4]; // 32 KB
  const int b = blockIdx.y;
  const unsigned tkey = thresh[b];
  const int tid = threadIdx.x;
  const int wave = tid >> 5, lane = tid & 31;
  const int wstride = P1_BLOCKS * WAVES;
  const unsigned long long gbase =
      (unsigned long long)(uintptr_t)logits + (unsigned long long)b * NPB * 4ull;
  unsigned char* mybuf = s_stage + wave * (2 * TILE_F * 4);
  const unsigned lbase = lds_off(mybuf);

  auto issue = [&](int t, int sel) {
    unsigned lo = lbase + (unsigned)sel * (TILE_F * 4) + (unsigned)lane * 16u;
    unsigned long long ga = gbase + (unsigned long long)t * (TILE_F * 4) + (unsigned)lane * 16u;
#pragma unroll
    for (int j = 0; j < IPT; ++j) { async_b128(lo, ga); lo += 512u; ga += 512ull; }
  };

  int t = blockIdx.x * WAVES + wave;
  int cur = 0;
  if (t < NTILES) issue(t, 0);
  for (; t < NTILES; t += wstride) {
    const int tn = t + wstride;
    if (tn < NTILES) { issue(tn, cur ^ 1); wait_async_half(); }
    else             { wait_async_zero(); }
    const float4* bp = reinterpret_cast<const float4*>(mybuf + cur * (TILE_F * 4));
#pragma unroll
    for (int k = 0; k < IPT; ++k) {
      float4 v = bp[k * 32 + lane];
      const int e0 = t * TILE_F + k * 128 + lane * 4;
      float vals[4] = {v.x, v.y, v.z, v.w};
#pragma unroll
      for (int m = 0; m < 4; ++m) {
        const int e = e0 + m;
        unsigned key = make_key(vals[m], obj, b, e);
        if (key >= tkey) {
          unsigned pos = atomicAdd(&ccnt[b], 1u);
          if (pos < (unsigned)CAP) {
            ckey[b * CAP + pos] = key;
            cidx[b * CAP + pos] = (unsigned)e;
          }
        }
      }
    }
    cur ^= 1;
  }
}

// ---------------- final: bitonic top-K + box post-process ----------------
__global__ __launch_bounds__(256) void k_final(const unsigned* __restrict__ ccnt,
                                               const unsigned* __restrict__ ckey,
                                               const unsigned* __restrict__ cidx,
                                               const float* __restrict__ boxes,
                                               const long long* __restrict__ tsz,
                                               float* __restrict__ out) {
  __shared__ unsigned sk[CAP];
  __shared__ unsigned si[CAP];
  const int b = blockIdx.x, tid = threadIdx.x;
  unsigned M = ccnt[b];
  if (M > (unsigned)CAP) M = (unsigned)CAP;
  for (int i = tid; i < CAP; i += 256) {
    if (i < (int)M) { sk[i] = ckey[b * CAP + i]; si[i] = cidx[b * CAP + i]; }
    else            { sk[i] = 0u;               si[i] = 0xFFFFFFFFu; }
  }
  __syncthreads();
  // bitonic sort: key descending, idx ascending tie-break (matches stable top_k)
  for (int size = 2; size <= CAP; size <<= 1) {
    for (int stride = size >> 1; stride > 0; stride >>= 1) {
      for (int p = tid; p < CAP / 2; p += 256) {
        const int i = 2 * stride * (p / stride) + (p % stride);
        const int j = i + stride;
        const bool dirDesc = ((i & size) == 0);
        unsigned ka = sk[i], kb = sk[j], ia = si[i], ib = si[j];
        const bool aFirst = (ka > kb) || (ka == kb && ia < ib);
        if (dirDesc != aFirst) { sk[i] = kb; sk[j] = ka; si[i] = ib; si[j] = ia; }
      }
      __syncthreads();
    }
  }
  if (tid < KTOP) {
    const unsigned key = sk[tid];
    unsigned idx = si[tid];
    float score;
    if (idx == 0xFFFFFFFFu) { idx = 0u; score = 0.0f; }
    else {
      const float x = unmono(key);
      score = 1.0f / (1.0f + __expf(-x));
    }
    const int qq  = (int)(idx / (unsigned)C_);
    const int cls = (int)(idx - (unsigned)qq * (unsigned)C_);
    const float* bx = boxes + ((size_t)b * Q_ + qq) * 4;
    const float cx = bx[0], cy = bx[1], w = bx[2], h = bx[3];
    const long long h0 = tsz[b * 2 + 0], w0 = tsz[b * 2 + 1];
    const float ms = (float)(h0 > w0 ? h0 : w0);
    float X0 = (cx - 0.5f * w) * ms, Y0 = (cy - 0.5f * h) * ms;
    float X1 = (cx + 0.5f * w) * ms, Y1 = (cy + 0.5f * h) * ms;
    const float LW = (float)w0, LH = (float)h0;
    X0 = fminf(fmaxf(X0, 0.0f), LW); Y0 = fminf(fmaxf(Y0, 0.0f), LH);
    X1 = fminf(fmaxf(X1, 0.0f), LW); Y1 = fminf(fmaxf(Y1, 0.0f), LH);
    const int r = b * KTOP + tid;
    out[r] = score;                       // scores
    out[B_ * KTOP + r] = (float)cls;      // labels
    float* ob = out + 2 * B_ * KTOP + (size_t)r * 4;
    ob[0] = X0; ob[1] = Y0; ob[2] = X1; ob[3] = Y1;
  }
}

__global__ void k_zero(unsigned* __restrict__ ws, int n) {
  const int i = blockIdx.x * blockDim.x + threadIdx.x;
  if (i < n) ws[i] = 0u;
}

// ---------------- launch ----------------
extern "C" void kernel_launch(void* const* d_in, const int* in_sizes, int n_in,
                              void* d_out, int out_size, void* d_ws, size_t ws_size,
                              hipStream_t stream) {
  (void)in_sizes; (void)n_in; (void)out_size; (void)ws_size;
  const float* logits     = (const float*)d_in[0];
  const float* obj        = (const float*)d_in[1];
  const float* boxes      = (const float*)d_in[2];
  const long long* tsz    = (const long long*)d_in[3];
  float* out = (float*)d_out;

  unsigned* ws     = (unsigned*)d_ws;
  unsigned* hist   = ws;                       // B*NBINS
  unsigned* thresh = hist + B_ * NBINS;        // B
  unsigned* ccnt   = thresh + B_;              // B
  unsigned* ckey   = ccnt + B_;                // B*CAP
  unsigned* cidx   = ckey + B_ * CAP;          // B*CAP

  const int nz = B_ * NBINS + 2 * B_;          // hist + thresh + counters
  k_zero<<<(nz + 255) / 256, 256, 0, stream>>>(ws, nz);

  dim3 gstream(P1_BLOCKS, B_);
  k_hist   <<<gstream, THREADS, 0, stream>>>(logits, obj, hist);
  k_thresh <<<B_,      256,     0, stream>>>(hist, thresh);
  k_collect<<<gstream, THREADS, 0, stream>>>(logits, obj, thresh, ccnt, ckey, cidx);
  k_final  <<<B_,      256,     0, stream>>>(ccnt, ckey, cidx, boxes, tsz, out);
}